// VIF_SSIM_Loss_38817914422008
// MI455X (gfx1250) — compile-verified
//
#include <hip/hip_runtime.h>
#include <hip/hip_bf16.h>

// Shapes from the reference: (32, 1, 704, 704) f32, 11x11 non-overlapping pooling.
#define W       704
#define KP      11
#define TXN     64          // tiles per row/col (704/11)
#define NB      32          // batch
#define NBLOCKS (NB * TXN)  // 2048 partial sums
#define NTILES  (NB * TXN * TXN) // 131072 tiles total

typedef __attribute__((ext_vector_type(4))) float f4;
typedef __attribute__((ext_vector_type(2))) float v2f;
typedef __attribute__((ext_vector_type(8))) float v8f;

// -------- Kernel 1: streaming per-tile-row statistics (HBM-bound phase) -----
// One block per (tile_row, batch). Threads 0..175 stream 11 rows x 704 cols of
// all three images with non-temporal b128 loads, accumulating 8 column stats.
// Threads 0..63 then form per-tile sums and the SSIM score; block emits one
// deterministic partial sum.
__global__ __launch_bounds__(256) void vif_ssim_tile_stats(
    const float* __restrict__ vis, const float* __restrict__ inf,
    const float* __restrict__ fus, float* __restrict__ partial)
{
    __shared__ float col[8][W];   // column sums: v,i,f,vv,ii,ff,vf,if
    __shared__ float red[TXN];

    const int ty = blockIdx.x;           // tile row 0..63
    const int b  = blockIdx.y;           // batch 0..31
    const size_t rowbase = (size_t)b * ((size_t)W * W) + (size_t)ty * KP * W;
    const int t = threadIdx.x;

    if (t < W / 4) {
        const f4* vp = (const f4*)(vis + rowbase) + t;
        const f4* ip = (const f4*)(inf + rowbase) + t;
        const f4* fp = (const f4*)(fus + rowbase) + t;
        f4 z = {0.0f, 0.0f, 0.0f, 0.0f};
        f4 sv = z, si = z, sf = z, svv = z, sii = z, sff = z, svf = z, sif = z;
        #pragma unroll
        for (int r = 0; r < KP; ++r) {
            f4 v = __builtin_nontemporal_load(vp + r * (W / 4));
            f4 i = __builtin_nontemporal_load(ip + r * (W / 4));
            f4 f = __builtin_nontemporal_load(fp + r * (W / 4));
            sv += v;      si += i;      sf += f;
            svv += v * v; sii += i * i; sff += f * f;
            svf += v * f; sif += i * f;
        }
        const int c = 4 * t;
        #pragma unroll
        for (int k = 0; k < 4; ++k) {
            col[0][c + k] = sv[k];  col[1][c + k] = si[k];
            col[2][c + k] = sf[k];  col[3][c + k] = svv[k];
            col[4][c + k] = sii[k]; col[5][c + k] = sff[k];
            col[6][c + k] = svf[k]; col[7][c + k] = sif[k];
        }
    }
    __syncthreads();

    if (t < TXN) {
        float s0 = 0.f, s1 = 0.f, s2 = 0.f, s3 = 0.f;
        float s4 = 0.f, s5 = 0.f, s6 = 0.f, s7 = 0.f;
        const int cb = t * KP;
        #pragma unroll
        for (int j = 0; j < KP; ++j) {
            s0 += col[0][cb + j]; s1 += col[1][cb + j];
            s2 += col[2][cb + j]; s3 += col[3][cb + j];
            s4 += col[4][cb + j]; s5 += col[5][cb + j];
            s6 += col[6][cb + j]; s7 += col[7][cb + j];
        }
        const float inv = 1.0f / (float)(KP * KP);
        const float C   = 0.0009f;
        const float vm = s0 * inv, im = s1 * inv, fm = s2 * inv;
        const float vvar = fabsf(s3 * inv - vm * vm);
        const float ivar = fabsf(s4 * inv - im * im);
        const float fvar = fabsf(s5 * inv - fm * fm);
        const float vcov = s6 * inv - vm * fm;
        const float icov = s7 * inv - im * fm;
        const float l_vf = (2.0f * vm * fm + C) / (vm * vm + fm * fm + C);
        const float l_if = (2.0f * im * fm + C) / (im * im + fm * fm + C);
        const float s_vf = (vcov + C) / (vvar + fvar + C);
        const float s_if = (icov + C) / (ivar + fvar + C);
        red[t] = (vm > im) ? (l_vf * s_vf) : (l_if * s_if);
    }
    __syncthreads();

    if (t == 0) {
        float acc = 0.0f;
        #pragma unroll
        for (int k = 0; k < TXN; ++k) acc += red[k];
        partial[b * gridDim.x + ty] = acc;
    }
}

// -------- Kernel 2: WMMA-based final reduction of the 2048 partials ---------
// One wave32. Each lane folds 2x32 partials into the two A-matrix VGPRs of a
// 16x4 f32 A tile. A x ones(4x16) via v_wmma_f32_16x16x4_f32 gives D rows =
// row sums of A; folding the 8 D VGPRs per lane plus the lane^16 half yields
// the full sum in f32 precision.
__global__ __launch_bounds__(32) void vif_ssim_wmma_reduce(
    const float* __restrict__ partial, float* __restrict__ out)
{
    const int lane = threadIdx.x;     // 0..31, EXEC all ones at the WMMA
    float a0 = 0.0f, a1 = 0.0f;
    const int c0 = (2 * lane) * 32;       // 64 slots x 32 values = 2048
    const int c1 = (2 * lane + 1) * 32;
    for (int k = 0; k < 32; ++k) {
        a0 += partial[c0 + k];
        a1 += partial[c1 + k];
    }
    v2f a;  a[0] = a0;   a[1] = a1;
    v2f bo; bo[0] = 1.0f; bo[1] = 1.0f;   // all-ones 4x16 B
    v8f c = {};
    // 8 args: (neg_a, A, neg_b, B, c_mod, C, reuse_a, reuse_b)
    v8f d = __builtin_amdgcn_wmma_f32_16x16x4_f32(
        false, a, false, bo, (short)0, c, false, false);
    float s = d[0] + d[1] + d[2] + d[3] + d[4] + d[5] + d[6] + d[7];
    float tot = s + __shfl_xor(s, 16, 32);  // combine M=0..7 and M=8..15 halves
    if (lane == 0) {
        out[0] = 1.0f - tot * (1.0f / (float)NTILES);
    }
}

extern "C" void kernel_launch(void* const* d_in, const int* in_sizes, int n_in,
                              void* d_out, int out_size, void* d_ws, size_t ws_size,
                              hipStream_t stream) {
    const float* vis = (const float*)d_in[0];
    const float* inf = (const float*)d_in[1];
    const float* fus = (const float*)d_in[2];
    float* out     = (float*)d_out;
    float* partial = (float*)d_ws;   // 2048 floats = 8 KB

    dim3 grid(TXN, NB);              // 64 tile-rows x 32 batches = 2048 blocks
    vif_ssim_tile_stats<<<grid, 256, 0, stream>>>(vis, inf, fus, partial);
    vif_ssim_wmma_reduce<<<1, 32, 0, stream>>>(partial, out);
}